// Ops_83700322664975
// MI455X (gfx1250) — compile-verified
//
#include <hip/hip_runtime.h>
#include <hip/hip_bf16.h>

// ---------------------------------------------------------------------------
// MI455X (gfx1250) fused keypoint-extraction pipeline.
//   K1 tile_mask_kernel : TDM tile DMA -> LDS, WMMA Hessian stencils,
//                         NMS/eof/edge/valid mask fusion, ballot compaction
//   K2 threshold        : suffix scan of 65536-bin float histogram
//   K3 compact          : gather candidates above threshold bin
//   K4 sort_emit        : single-WG bitonic sort (8192 slots in LDS), emit
// ---------------------------------------------------------------------------

typedef float  v2f __attribute__((ext_vector_type(2)));
typedef float  v8f __attribute__((ext_vector_type(8)));
typedef unsigned int v4u __attribute__((ext_vector_type(4)));
typedef int    v4i __attribute__((ext_vector_type(4)));
typedef int    v8i __attribute__((ext_vector_type(8)));

#define HW        4096
#define TILE      128
#define PITCH     132              // 128 core + pad-4 (132 % 64 banks = 4 -> conflict-free columns)
#define SLABROWS  130
#define CCAP      (1u << 22)       // candidate capacity (4M records)
#define FCAP      8192             // final sort capacity
#define HBINS     65536

// workspace layout (bytes)
#define WS_CTRL       0            // [0]=cand_count [1]=final_count [2]=thr_bin
#define WS_HIST       32
#define WS_CSCORE     (32 + HBINS * 4)                 // float[CCAP]
#define WS_CIDX       (WS_CSCORE + CCAP * 4)           // uint [CCAP]
#define WS_FSCORE     (WS_CIDX   + CCAP * 4)           // float[FCAP]
#define WS_FIDX       (WS_FSCORE + FCAP * 4)           // uint [FCAP]

__global__ void init_kernel(unsigned* ctrl_hist, int nwords) {
  int i = blockIdx.x * blockDim.x + threadIdx.x;
  if (i < nwords) ctrl_hist[i] = 0u;
}

// ---------------------------------------------------------------------------
// K1: one 128x128 tile per 256-thread workgroup (8 wave32).
// ---------------------------------------------------------------------------
__global__ __launch_bounds__(256) void tile_mask_kernel(
    const float* __restrict__ score, const unsigned char* __restrict__ valid,
    const int* __restrict__ score_thld_p, const int* __restrict__ edge_thld_p,
    const int* __restrict__ eof_p,
    unsigned* __restrict__ cand_count, float* __restrict__ cand_score,
    unsigned* __restrict__ cand_idx, unsigned* __restrict__ hist) {

  __shared__ float slab[SLABROWS * PITCH];   // only shared object -> LDS offset 0

  const int gr0 = blockIdx.y * TILE;
  const int gc0 = blockIdx.x * TILE;

  // ---- TDM: DMA the 128x128 core into slab[1..128][1..128] (pitch 132) ----
  if (threadIdx.x < 32) {
    unsigned long long gaddr =
        (unsigned long long)(const void*)score +
        (((unsigned long long)gr0 * HW + (unsigned long long)gc0) << 2);
    v4u g0;
    g0.x = 1u;                                       // count=1, user descriptor
    g0.y = (unsigned)((1 * PITCH + 1) * 4);          // LDS byte addr of core start
    g0.z = (unsigned)(gaddr & 0xFFFFFFFFull);        // global_addr[31:0]
    g0.w = (unsigned)((gaddr >> 32) & 0x01FFFFFFull) // global_addr[56:32]
         | (2u << 30);                               // type = image
    v8i g1;
    g1[0] = (int)((2u << 16)      // data_size = 4B
                | (1u << 20)      // pad_enable
                | (6u << 22)      // pad_interval: 128 dwords (512B)
                | (3u << 25));    // pad_amount: 4 dwords -> pitch 132
    g1[1] = (int)((unsigned)HW << 16);               // tensor_dim0 lo16 @ [31:16]
    g1[2] = (int)(((unsigned)HW >> 16) | ((unsigned)HW << 16)); // dim0 hi | dim1 lo
    g1[3] = (int)(((unsigned)HW >> 16) | ((unsigned)TILE << 16)); // dim1 hi | tile_dim0
    g1[4] = (int)(unsigned)TILE;                     // tile_dim1=128, tile_dim2=0
    g1[5] = (int)(unsigned)HW;                       // tensor_dim0_stride lo32
    g1[6] = 0;                                       // stride hi | dim1_stride lo
    g1[7] = 0;
    v4i z4 = {0, 0, 0, 0};
#if defined(__clang_major__) && (__clang_major__ >= 23)
    v8i z8 = {0, 0, 0, 0, 0, 0, 0, 0};
    __builtin_amdgcn_tensor_load_to_lds(g0, g1, z4, z4, z8, 0);
#else
    __builtin_amdgcn_tensor_load_to_lds(g0, g1, z4, z4, 0);
#endif
  }

  // ---- halo ring (516 elems) with guarded VMEM loads, overlapped with TDM --
  for (int i = threadIdx.x; i < 2 * SLABROWS + 2 * TILE; i += 256) {
    int r, c;
    if (i < SLABROWS)            { r = 0;              c = i; }
    else if (i < 2 * SLABROWS)   { r = SLABROWS - 1;   c = i - SLABROWS; }
    else if (i < 2 * SLABROWS + TILE) { r = i - 2 * SLABROWS + 1; c = 0; }
    else                         { r = i - 2 * SLABROWS - TILE + 1; c = PITCH - 3; } // col 129
    int grow = gr0 - 1 + r, gcol = gc0 - 1 + c;
    float v = 0.f;
    if ((unsigned)grow < (unsigned)HW && (unsigned)gcol < (unsigned)HW)
      v = score[(size_t)grow * HW + gcol];
    slab[r * PITCH + c] = v;
  }

  if (threadIdx.x < 32) __builtin_amdgcn_s_wait_tensorcnt(0);
  __syncthreads();

  // ---- per-wave 16x16 subtiles: WMMA Hessian stencils + masks -------------
  const int lane = threadIdx.x & 31;
  const int wave = threadIdx.x >> 5;
  const int n    = lane & 15;
  const int hi   = lane >> 4;                    // 0 or 1 (C-frag M offset 8)

  const float sthld = (float)(*score_thld_p);
  const int   eof   = *eof_p;
  float thld_e;
  { float et = (float)(*edge_thld_p); thld_e = (et + 1.f) * (et + 1.f) / et; }

  for (int st = wave; st < 64; st += 8) {
    const int sr = (st >> 3) * 16;
    const int sc = (st & 7) * 16;

    v8f dii = {};
    v8f djj = {};
    // dii = A_band(16x18) * X(18x16);  djj = X(16x18) * B_band(18x16). K padded to 20.
    // Branchless: for kk >= 18 the band coefficient is exactly 0 (bands reach
    // at most index 17), so the X operand is address-clamped, never zeroed.
    for (int c4 = 0; c4 < 5; ++c4) {
      const int kbase = c4 * 4 + 2 * hi;
      v2f a, b;
#pragma unroll
      for (int j = 0; j < 2; ++j) {               // vertical stencil (dii)
        const int kk = kbase + j;
        const int m  = lane & 15;
        const int kc = kk < 18 ? kk : 17;         // address clamp only
        a[j] = (kk == m ? 1.f : 0.f) + (kk == m + 1 ? -2.f : 0.f) + (kk == m + 2 ? 1.f : 0.f);
        b[j] = slab[(sr + kc) * PITCH + (sc + n + 1)];
      }
      dii = __builtin_amdgcn_wmma_f32_16x16x4_f32(false, a, false, b,
                                                  (short)0, dii, false, false);
#pragma unroll
      for (int j = 0; j < 2; ++j) {               // horizontal stencil (djj)
        const int kk = kbase + j;
        const int m  = lane & 15;
        const int kc = kk < 18 ? kk : 17;         // address clamp only
        a[j] = slab[(sr + m + 1) * PITCH + (sc + kc)];
        b[j] = (kk == n ? 1.f : 0.f) + (kk == n + 1 ? -2.f : 0.f) + (kk == n + 2 ? 1.f : 0.f);
      }
      djj = __builtin_amdgcn_wmma_f32_16x16x4_f32(false, a, false, b,
                                                  (short)0, djj, false, false);
    }

#pragma unroll
    for (int r = 0; r < 8; ++r) {
      const int m    = r + (hi ? 8 : 0);
      const int grow = gr0 + sr + m;
      const int gcol = gc0 + sc + n;
      const float* p = &slab[(sr + m + 1) * PITCH + (sc + n + 1)];
      const float s = p[0];
      // 3x3 NMS max (corners shared with dij below -> CSE'd)
      float mx = fmaxf(fmaxf(p[-PITCH - 1], p[-PITCH]), p[-PITCH + 1]);
      mx = fmaxf(mx, fmaxf(p[-1], fmaxf(s, p[1])));
      mx = fmaxf(mx, fmaxf(p[PITCH - 1], fmaxf(p[PITCH], p[PITCH + 1])));
      // Hessian edge test
      const float diiv = dii[r], djjv = djj[r];
      const float dijv = 0.25f * (p[-PITCH - 1] - p[-PITCH + 1] - p[PITCH - 1] + p[PITCH + 1]);
      const float det  = diiv * djjv - dijv * dijv;
      const float tr   = diiv + djjv;

      bool ok = (s > sthld) && (s == mx);
      ok = ok && (grow >= eof) && (grow < HW - eof) && (gcol >= eof) && (gcol < HW - eof);
      ok = ok && (det > 0.f) && (tr * tr <= thld_e * det);
      ok = ok && (valid[(size_t)grow * HW + gcol] != 0);

      // wave32 ballot compaction: one atomic per wave per row
      unsigned b32 = (unsigned)__ballot(ok);
      int cnt = __popc(b32);
      if (cnt) {
        int leader = __ffs(b32) - 1;
        unsigned base = 0;
        if (lane == leader) base = atomicAdd(cand_count, (unsigned)cnt);
        base = (unsigned)__shfl((int)base, leader);
        if (ok) {
          unsigned off = base + (unsigned)__popc(b32 & ((1u << lane) - 1u));
          if (off < CCAP) {
            cand_score[off] = s;
            cand_idx[off]   = (unsigned)grow * HW + (unsigned)gcol;
            atomicAdd(&hist[(__float_as_uint(s) >> 15) & 0xFFFFu], 1u);
          }
        }
      }
    }
  }
}

// ---------------------------------------------------------------------------
// K2: suffix-scan histogram (positive-float bits are monotone) -> threshold bin
//     256 threads reduce 256-bin chunks; one lane walks <=512 entries.
// ---------------------------------------------------------------------------
__global__ __launch_bounds__(256) void threshold_kernel(
    const unsigned* __restrict__ hist, unsigned* __restrict__ ctrl, int k) {
  __shared__ unsigned chunk[256];
  const int c = threadIdx.x;
  unsigned s = 0;
  for (int i = 0; i < 256; ++i) s += hist[c * 256 + i];
  chunk[c] = s;
  __syncthreads();
  if (c == 0) {
    unsigned cum = 0;
    int ch = 255;
    for (; ch >= 0; --ch) {
      if (cum + chunk[ch] >= (unsigned)k) break;
      cum += chunk[ch];
    }
    int bin = 0;
    if (ch >= 0) {
      int b = ch * 256 + 255;
      for (; b >= ch * 256; --b) {
        cum += hist[b];
        if (cum >= (unsigned)k) break;
      }
      bin = (b < ch * 256) ? ch * 256 : b;
    }
    ctrl[2] = (unsigned)bin;
  }
}

// ---------------------------------------------------------------------------
// K3: compact candidates in bins >= threshold into the final pool
// ---------------------------------------------------------------------------
__global__ __launch_bounds__(256) void compact_kernel(
    const float* __restrict__ cs, const unsigned* __restrict__ ci,
    const unsigned* __restrict__ ctrl, unsigned* __restrict__ final_count,
    float* __restrict__ fs, unsigned* __restrict__ fi) {
  unsigned ncand = ctrl[0];
  if (ncand > CCAP) ncand = CCAP;
  const unsigned thr = ctrl[2];
  const unsigned stride = gridDim.x * blockDim.x;
  for (unsigned i = blockIdx.x * blockDim.x + threadIdx.x; i < ncand; i += stride) {
    const float s = cs[i];
    const unsigned bin = (__float_as_uint(s) >> 15) & 0xFFFFu;
    if (bin >= thr) {
      unsigned off = atomicAdd(final_count, 1u);
      if (off < FCAP) { fs[off] = s; fi[off] = ci[i]; }
    }
  }
}

// ---------------------------------------------------------------------------
// K4: single-WG bitonic sort (descending) over <=8192 survivors in LDS, emit
//     out[0..k)=rows, out[k..2k)=cols, out[2k..3k)=scores
// ---------------------------------------------------------------------------
__global__ __launch_bounds__(1024) void sort_emit_kernel(
    const float* __restrict__ fs, const unsigned* __restrict__ fi,
    const unsigned* __restrict__ final_count, float* __restrict__ out, int k) {
  __shared__ float    key[FCAP];
  __shared__ unsigned val[FCAP];
  unsigned nf = *final_count;
  if (nf > FCAP) nf = FCAP;

  for (int i = threadIdx.x; i < FCAP; i += 1024) {
    key[i] = (i < (int)nf) ? fs[i] : -__builtin_inff();
    val[i] = (i < (int)nf) ? fi[i] : 0u;
  }
  __syncthreads();

  for (int size = 2; size <= FCAP; size <<= 1) {
    for (int stride = size >> 1; stride > 0; stride >>= 1) {
      for (int i = threadIdx.x; i < FCAP / 2; i += 1024) {
        const int lo = (i / stride) * (stride << 1) + (i % stride);
        const int hiidx = lo + stride;
        const bool desc = ((lo & size) == 0);
        const float a = key[lo], b = key[hiidx];
        if ((a < b) == desc) {
          key[lo] = b; key[hiidx] = a;
          const unsigned t = val[lo]; val[lo] = val[hiidx]; val[hiidx] = t;
        }
      }
      __syncthreads();
    }
  }

  for (int t = threadIdx.x; t < k; t += 1024) {
    float s = -__builtin_inff();
    unsigned id = 0u;
    if (t < FCAP) { s = key[t]; id = val[t]; }
    out[t]         = (float)(id >> 12);      // row = id / 4096
    out[k + t]     = (float)(id & 4095u);    // col = id % 4096
    out[2 * k + t] = s;
  }
}

// ---------------------------------------------------------------------------
extern "C" void kernel_launch(void* const* d_in, const int* in_sizes, int n_in,
                              void* d_out, int out_size, void* d_ws, size_t ws_size,
                              hipStream_t stream) {
  const float*         score = (const float*)d_in[0];
  const unsigned char* valid = (const unsigned char*)d_in[1];
  const int* score_thld_p = (const int*)d_in[3];
  const int* edge_thld_p  = (const int*)d_in[4];
  const int* eof_p        = (const int*)d_in[6];
  const int  k            = out_size / 3;   // outputs: rows[k], cols[k], scores[k]

  unsigned char* ws = (unsigned char*)d_ws;
  unsigned* ctrl = (unsigned*)(ws + WS_CTRL);      // [0]=cand_count [1]=final_count [2]=thr
  unsigned* hist = (unsigned*)(ws + WS_HIST);
  float*    cs   = (float*)(ws + WS_CSCORE);
  unsigned* ci   = (unsigned*)(ws + WS_CIDX);
  float*    fsd  = (float*)(ws + WS_FSCORE);
  unsigned* fid  = (unsigned*)(ws + WS_FIDX);

  const int nwords = 8 + HBINS;
  init_kernel<<<(nwords + 255) / 256, 256, 0, stream>>>(ctrl, nwords);

  dim3 grid(HW / TILE, HW / TILE);
  tile_mask_kernel<<<grid, 256, 0, stream>>>(score, valid, score_thld_p,
                                             edge_thld_p, eof_p,
                                             &ctrl[0], cs, ci, hist);

  threshold_kernel<<<1, 256, 0, stream>>>(hist, ctrl, k);
  compact_kernel<<<1024, 256, 0, stream>>>(cs, ci, ctrl, &ctrl[1], fsd, fid);
  sort_emit_kernel<<<1, 1024, 0, stream>>>(fsd, fid, &ctrl[1], (float*)d_out, k);
}